// BilateralSlice_9405978378861
// MI455X (gfx1250) — compile-verified
//
#include <hip/hip_runtime.h>

// CDNA5 / gfx1250 bilateral slice using V_WMMA_F32_16X16X4_F32.
// Per grid cell: D(16ch x 16px) = A(16x32 grid footprint) x B(32x16 trilinear
// weights), K=32 as 8 chained 16x16x4 f32 WMMAs. Grid footprint is constant
// per block; weights are built directly in the WMMA B-register layout (both
// lane halves track the same 16 pixels, so no LDS transpose is needed).
// Depth factors use the closed-form hat function max(0, 1-|dcc-d|), removing
// the integer floor/compare/select chain entirely.

typedef float v2f __attribute__((ext_vector_type(2)));
typedef float v8f __attribute__((ext_vector_type(8)));

#define NCELL 15   // 16 grid nodes -> 15 cells/axis; pixel 1023 folded into cell 14 (fh=1)

__global__ __launch_bounds__(256)
void bslice_wmma_kernel(const float* __restrict__ bgrid,
                        const float* __restrict__ guide,
                        float* __restrict__ out)
{
    const int lane = threadIdx.x & 31;
    const int wave = threadIdx.x >> 5;
    const int half = lane >> 4;        // which 16-lane half of the wave
    const int lx   = lane & 15;

    int idx = blockIdx.x;
    const int r     = idx & 3;  idx >>= 2;       // row split within the cell
    const int cellI = idx % NCELL; idx /= NCELL; // i axis (H) -> Wg index
    const int cellJ = idx % NCELL;               // j axis (W) -> Hg index
    const int b     = idx / NCELL;

    // pixel ranges of this cell: ceil(k*1023/15) .. next-1
    const int sJ = (cellJ * 1023 + 14) / 15;
    const int eJ = (cellJ == NCELL - 1) ? 1023 : ((cellJ + 1) * 1023 + 14) / 15 - 1;
    const int sI = (cellI * 1023 + 14) / 15;
    const int eI = (cellI == NCELL - 1) ? 1023 : ((cellI + 1) * 1023 + 14) / 15 - 1;

    const int cnt   = eI - sI + 1;
    const int per   = (cnt + 3) >> 2;
    const int rowLo = sI + r * per;
    const int rowHi = min(rowLo + per - 1, eI);

    // ---- A fragments: grid footprint (16 ch x K=32), constant for the block.
    // A-layout (f32 16x4): lane m holds M=m%16; VGPR0 -> K = 4q + (half?2:0),
    // VGPR1 -> K+1.  k = (hbit*2+wbit)*8 + d.
    const int ch   = lx;
    const int chc  = min(ch, 11);
    const int dOff = half ? 2 : 0;
    const float* gbase = bgrid + (size_t)((b * 12 + chc) * 16) * (16 * 8);
    v2f afrag[8];
    #pragma unroll
    for (int q = 0; q < 8; ++q) {
        const int hb = q >> 2;
        const int wb = (q >> 1) & 1;
        const int d0 = ((q & 1) << 2) + dOff;            // 0/2/4/6 -> 8B aligned
        const float* p = gbase + ((cellJ + hb) * 16 + (cellI + wb)) * 8 + d0;
        v2f a = *(const v2f*)p;
        if (ch >= 12) { a.x = 0.0f; a.y = 0.0f; }        // zero-pad channels 12..15
        afrag[q] = a;
    }

    const float* gpix  = guide + (size_t)b * (1024u * 1024u);
    float*       obase = out   + (size_t)b * 12u * 1024u * 1024u;

    // Uniform per-channel-register bases; register v holds channel v (lanes
    // 0-15) and channel v+8 (lanes 16-31), the +8 part folded into voffset.
    float* oc[8];
    #pragma unroll
    for (int v = 0; v < 8; ++v)
        oc[v] = obase + ((unsigned)v << 20);

    const float cellJf = (float)cellJ;
    const float dOfff  = (float)dOff;                    // 0.0 or 2.0 per half

    for (int i = rowLo + wave; i <= rowHi; i += 8) {
        const float fw  = ((float)i * (1.0f / 1023.0f)) * 15.0f - (float)cellI;
        const float fw0 = 1.0f - fw;
        const unsigned rowoff = (unsigned)i << 10;       // i*1024 (elements)

        // Preload guide for all 5 pixel-groups of this row (hide load latency).
        float gg[5];
        #pragma unroll
        for (int t = 0; t < 5; ++t) {
            int jj = sJ + (t << 4) + lx;
            if (t == 4) jj = min(jj, 1023);              // only the tail can run off-image
            gg[t] = gpix[rowoff + (unsigned)jj];
        }

        #pragma unroll
        for (int t = 0; t < 5; ++t) {
            const int   j = sJ + (t << 4) + lx;
            const float g = gg[t];

            // Depth factors: trilinear hat max(0, 1-|dcc-d|) on the 4 slots
            // this lane touches (dOff, dOff+1, dOff+4, dOff+5).
            const float dcc = fminf(fmaxf(g * 7.0f, 0.0f), 7.0f);
            const float fA = fmaxf(1.0f - fabsf(dcc - dOfff), 0.0f);
            const float fB = fmaxf(1.0f - fabsf(dcc - (dOfff + 1.0f)), 0.0f);
            const float fC = fmaxf(1.0f - fabsf(dcc - (dOfff + 4.0f)), 0.0f);
            const float fD = fmaxf(1.0f - fabsf(dcc - (dOfff + 5.0f)), 0.0f);

            const float fh  = ((float)j * (1.0f / 1023.0f)) * 15.0f - cellJf;
            const float fh0 = 1.0f - fh;
            const float hw[4] = { fh0 * fw0, fh0 * fw, fh * fw0, fh * fw };

            v8f acc = {0.f, 0.f, 0.f, 0.f, 0.f, 0.f, 0.f, 0.f};
            #pragma unroll
            for (int q = 0; q < 8; ++q) {
                const float c = hw[q >> 1];
                v2f bf;
                bf.x = c * ((q & 1) ? fC : fA);
                bf.y = c * ((q & 1) ? fD : fB);
                // (neg_a, A, neg_b, B, c_mod, C, reuse_a, reuse_b)
                acc = __builtin_amdgcn_wmma_f32_16x16x4_f32(
                        false, afrag[q], false, bf, (short)0, acc, false, false);
            }

            // D layout: VGPR v -> channel v + 8*half, pixel column lx.
            const int voff = (half ? (8 << 20) : 0) + (int)rowoff + j;

            if (t < 4) {                                  // full group: no j mask
                #pragma unroll
                for (int v = 0; v < 4; ++v)               // channels v / v+8: all lanes
                    __builtin_nontemporal_store(acc[v], oc[v] + voff);
                if (!half) {                              // channels 4..7: lower half only
                    #pragma unroll
                    for (int v = 4; v < 8; ++v)
                        __builtin_nontemporal_store(acc[v], oc[v] + voff);
                }
            } else {                                      // tail group: mask j <= eJ
                const bool jok = (j <= eJ);
                #pragma unroll
                for (int v = 0; v < 4; ++v)
                    if (jok)
                        __builtin_nontemporal_store(acc[v], oc[v] + voff);
                if (!half && jok) {
                    #pragma unroll
                    for (int v = 4; v < 8; ++v)
                        __builtin_nontemporal_store(acc[v], oc[v] + voff);
                }
            }
        }
    }
}

extern "C" void kernel_launch(void* const* d_in, const int* in_sizes, int n_in,
                              void* d_out, int out_size, void* d_ws, size_t ws_size,
                              hipStream_t stream) {
    const float* grid_p  = (const float*)d_in[0];   // (4,12,16,16,8) f32
    const float* guide_p = (const float*)d_in[1];   // (4,1,1024,1024) f32
    // d_in[2] (input_image) is unused by the reference math; only shapes H,W.
    float* out_p = (float*)d_out;                   // (4,12,1024,1024) f32

    const int nblocks = 4 * NCELL * NCELL * 4;      // b x cellJ x cellI x rowsplit
    hipLaunchKernelGGL(bslice_wmma_kernel, dim3(nblocks), dim3(256), 0, stream,
                       grid_p, guide_p, out_p);
}